// GCNEncoder_78709570666587
// MI455X (gfx1250) — compile-verified
//
#include <hip/hip_runtime.h>

#define N_NODES 100000
#define N_EDGES 1280000
#define DIM 64
#define N_STRIPS (N_NODES / 16)   // 6250, exact
#define PSTRIDE 160               // floats per k-pair row in LDS (padded: 160 % 128 != 0,
                                  // 160 dwords ≡ 32 mod 64 banks -> khalf=1 lanes use banks 32..63)

typedef float v2f __attribute__((ext_vector_type(2)));
typedef float v8f __attribute__((ext_vector_type(8)));

// ---------------------------------------------------------------- zero fill
__global__ void gcn_zero4(float4* __restrict__ p, int n4) {
    int i = blockIdx.x * blockDim.x + threadIdx.x;
    if (i < n4) p[i] = make_float4(0.f, 0.f, 0.f, 0.f);
}

// ------------------------------------------------- edge scatter: agg += w*x[src]
// 16 threads per edge, each handling a float4 (4 channels) -> the 256B row of
// x[src] is read as one contiguous coalesced segment; scatter uses native
// global_atomic_add_f32 (L2 atomics; agg stays resident in the 192MB L2).
__global__ __launch_bounds__(256) void gcn_scatter(
    const float* __restrict__ x,
    const int*   __restrict__ src,
    const int*   __restrict__ dst,
    const float* __restrict__ w,
    float*       __restrict__ agg) {
    long long t = (long long)blockIdx.x * blockDim.x + threadIdx.x;
    int e = (int)(t >> 4);
    if (e >= N_EDGES) return;
    int c = ((int)t & 15) << 2;

    int   s  = src[e];
    int   d  = dst[e];
    float wt = w[e];

    float4 xv = *(const float4*)(x + (long long)s * DIM + c);
    float* ap = agg + (long long)d * DIM + c;
    unsafeAtomicAdd(ap + 0, wt * xv.x);
    unsafeAtomicAdd(ap + 1, wt * xv.y);
    unsafeAtomicAdd(ap + 2, wt * xv.z);
    unsafeAtomicAdd(ap + 3, wt * xv.w);
}

// -------------------------------------- fused dual-GEMM + bias (+ReLU) via
// V_WMMA_F32_16X16X4_F32: out = agg@Wrel + b + xin@Wroot, one wave per
// 16-row strip, 4 column tiles of 16, K=64 in 16 chunks of 4.
// Weights live in LDS in a fragment-native paired layout so each B fragment
// is a single conflict-free ds_load_b64:
//   sW[p*PSTRIDE + n*2 + 0] = W[2p  ][n]
//   sW[p*PSTRIDE + n*2 + 1] = W[2p+1][n]
template <bool RELU>
__global__ __launch_bounds__(256) void gcn_gemm(
    const float* __restrict__ agg,    // [N,64]
    const float* xin,                 // [N,64] (may alias out)
    const float* __restrict__ Wrel,   // [64,64] row-major K x N
    const float* __restrict__ brel,   // [64]
    const float* __restrict__ Wroot,  // [64,64]
    float* out) {                     // [N,64] (may alias xin)
    __shared__ float sWrel[32 * PSTRIDE];
    __shared__ float sWroot[32 * PSTRIDE];
    __shared__ float sb[DIM];

    // stage both weight matrices + bias in LDS, reshuffling into pair layout
    for (int i = threadIdx.x; i < DIM * DIM / 4; i += 256) {
        int k  = (i * 4) / DIM;
        int n0 = (i * 4) % DIM;
        int base = (k >> 1) * PSTRIDE + (k & 1);
        float4 r = ((const float4*)Wrel)[i];
        float4 t = ((const float4*)Wroot)[i];
        sWrel[base + (n0 + 0) * 2] = r.x;
        sWrel[base + (n0 + 1) * 2] = r.y;
        sWrel[base + (n0 + 2) * 2] = r.z;
        sWrel[base + (n0 + 3) * 2] = r.w;
        sWroot[base + (n0 + 0) * 2] = t.x;
        sWroot[base + (n0 + 1) * 2] = t.y;
        sWroot[base + (n0 + 2) * 2] = t.z;
        sWroot[base + (n0 + 3) * 2] = t.w;
    }
    if (threadIdx.x < DIM) sb[threadIdx.x] = brel[threadIdx.x];
    __syncthreads();

    int wave  = threadIdx.x >> 5;
    int lane  = threadIdx.x & 31;
    int strip = blockIdx.x * 8 + wave;
    if (strip >= N_STRIPS) return;           // wave-uniform: EXEC stays all-1s

    int row0  = strip * 16;
    int m     = lane & 15;                   // A row / B,C column within tile
    int khalf = lane >> 4;                   // 0: K=kb,kb+1   1: K=kb+2,kb+3

    v8f acc[4];
#pragma unroll
    for (int nt = 0; nt < 4; ++nt) acc[nt] = (v8f){0, 0, 0, 0, 0, 0, 0, 0};

    const float* aRow = agg + (long long)(row0 + m) * DIM;
    const float* xRow = xin + (long long)(row0 + m) * DIM;
    // per-lane LDS base: column term n*2 = (nt*16 + m)*2; nt handled by imm offset
    const float* bRel0  = sWrel  + khalf * PSTRIDE + m * 2;
    const float* bRoot0 = sWroot + khalf * PSTRIDE + m * 2;

#pragma unroll 4
    for (int ks = 0; ks < 16; ++ks) {
        int kb = ks * 4 + khalf * 2;
        // A fragments (16x4 f32 layout): lane holds {A[m][kb], A[m][kb+1]}
        v2f aA = *(const v2f*)(aRow + kb);
        v2f aX = *(const v2f*)(xRow + kb);
        const float* bR = bRel0  + ks * 2 * PSTRIDE;
        const float* bX = bRoot0 + ks * 2 * PSTRIDE;
#pragma unroll
        for (int nt = 0; nt < 4; ++nt) {
            // B fragments: single aligned ds_load_b64, no repacking
            v2f fR = *(const v2f*)(bR + nt * 32);
            v2f fX = *(const v2f*)(bX + nt * 32);
            acc[nt] = __builtin_amdgcn_wmma_f32_16x16x4_f32(
                false, aA, false, fR, (short)0, acc[nt], false, false);
            acc[nt] = __builtin_amdgcn_wmma_f32_16x16x4_f32(
                false, aX, false, fX, (short)0, acc[nt], false, false);
        }
    }

    // C/D layout: VGPR j -> row (j + khalf*8), column m of the tile
#pragma unroll
    for (int nt = 0; nt < 4; ++nt) {
        int   n    = nt * 16 + m;
        float bias = sb[n];
#pragma unroll
        for (int j = 0; j < 8; ++j) {
            int   row = row0 + j + khalf * 8;
            float v   = acc[nt][j] + bias;
            if (RELU) v = v > 0.f ? v : 0.f;
            out[(long long)row * DIM + n] = v;
        }
    }
}

extern "C" void kernel_launch(void* const* d_in, const int* in_sizes, int n_in,
                              void* d_out, int out_size, void* d_ws, size_t ws_size,
                              hipStream_t stream) {
    const float* x   = (const float*)d_in[0];
    const int*   ei  = (const int*)d_in[1];
    const float* ew  = (const float*)d_in[2];
    const float* Wrel1  = (const float*)d_in[4];
    const float* brel1  = (const float*)d_in[5];
    const float* Wroot1 = (const float*)d_in[6];
    const float* Wrel2  = (const float*)d_in[7];
    const float* brel2  = (const float*)d_in[8];
    const float* Wroot2 = (const float*)d_in[9];
    const float* Wrel3  = (const float*)d_in[10];
    const float* brel3  = (const float*)d_in[11];
    const float* Wroot3 = (const float*)d_in[12];

    const int* src = ei;            // edge_index row 0
    const int* dst = ei + N_EDGES;  // edge_index row 1

    float* agg = (float*)d_ws;                 // [N,64]
    float* h   = agg + (size_t)N_NODES * DIM;  // [N,64]
    float* o   = (float*)d_out;

    const int nElem      = N_NODES * DIM;
    const int zeroBlocks = (nElem / 4 + 255) / 256;
    const int scatBlocks = (int)(((long long)N_EDGES * 16 + 255) / 256);
    const int gemmBlocks = (N_STRIPS + 7) / 8;

    // ---- layer 1: h = relu(agg(x) @ Wrel1 + b1 + x @ Wroot1)
    gcn_zero4<<<zeroBlocks, 256, 0, stream>>>((float4*)agg, nElem / 4);
    gcn_scatter<<<scatBlocks, 256, 0, stream>>>(x, src, dst, ew, agg);
    gcn_gemm<true><<<gemmBlocks, 256, 0, stream>>>(agg, x, Wrel1, brel1, Wroot1, h);

    // ---- layer 2: h = relu(agg(h) @ Wrel2 + b2 + h @ Wroot2)   (in-place safe)
    gcn_zero4<<<zeroBlocks, 256, 0, stream>>>((float4*)agg, nElem / 4);
    gcn_scatter<<<scatBlocks, 256, 0, stream>>>(h, src, dst, ew, agg);
    gcn_gemm<true><<<gemmBlocks, 256, 0, stream>>>(agg, h, Wrel2, brel2, Wroot2, h);

    // ---- layer 3: out = agg(h) @ Wrel3 + b3 + h @ Wroot3   (no relu)
    gcn_zero4<<<zeroBlocks, 256, 0, stream>>>((float4*)agg, nElem / 4);
    gcn_scatter<<<scatBlocks, 256, 0, stream>>>(h, src, dst, ew, agg);
    gcn_gemm<false><<<gemmBlocks, 256, 0, stream>>>(agg, h, Wrel3, brel3, Wroot3, o);
}